// CTCLoss_51505247814330
// MI455X (gfx1250) — compile-verified
//
#include <hip/hip_runtime.h>
#include <hip/hip_bf16.h>
#include <stdint.h>
#include <math.h>

// CTC mean loss for B=256,T=1024,V=64,L=64 (S=129).
// Design (MI455X / gfx1250, wave32):
//  - 1 workgroup (160 thr = 5 waves) per batch element; whole DP in LDS.
//  - logits staged per 16-step tile with global_load_async_to_lds_b128
//    (ASYNCcnt, double buffered) -> loads overlap the sequential DP.
//  - em gather (logp at extended labels) done as an exact one-hot matmul
//    on the matrix pipe: V_WMMA_F32_16X16X4_F32, 10 s-tiles (incl. 1 dead
//    all-zero tile so every wave runs both chains with full EXEC), 16 K-steps.
//  - logaddexp recurrence: ping-pong alpha in LDS, 1 barrier/step.
//  - per-example losses -> d_ws; deterministic tree reduce -> mean.

#define NEGV    (-1e30f)
#define S_EXT   129
#define LMAX    64
#define VDIM    64
#define TILE_T  16
#define ROWP    68        // padded LDS row stride (floats) for staged logits
#define EMP     145       // padded row stride (floats) for em tile (odd: no bank conflict)
#define NTHR    160

typedef float v2f __attribute__((ext_vector_type(2)));
typedef float v8f __attribute__((ext_vector_type(8)));

__global__ __launch_bounds__(NTHR)
void ctc_dp_kernel(const int* __restrict__ labels,   // [B, LMAX] (int32 view)
                   const float* __restrict__ logits, // [B, T, VDIM] fp32
                   const int* __restrict__ lmask,    // [B, T] 0/1
                   float* __restrict__ loss_out,     // [B]
                   int T)
{
    const int b    = blockIdx.x;
    const int tid  = threadIdx.x;
    const int lane = tid & 31;
    const int wave = tid >> 5;

    __shared__ __align__(16) float ring[2][TILE_T * ROWP]; // staged logits (async dst)
    __shared__ float em[TILE_T * EMP];         // em[tt][s] for current tile
    __shared__ float lseRow[TILE_T];
    __shared__ float alphaBuf[2][S_EXT + 3];   // [0..1] = NEG pads, state s at +2
    __shared__ int   extLab[NTHR];             // blank-interleaved labels (pad -1)
    __shared__ int   allowS[S_EXT];
    __shared__ int   redA[NTHR], redB[NTHR];
    __shared__ int   lens[2];

    // ---- setup: extended labels, lengths, skip mask ----
    int es = -1;
    if (tid < S_EXT) es = (tid & 1) ? labels[b * LMAX + (tid >> 1)] : 0;
    extLab[tid] = es;

    int msum = 0;
    for (int t = tid; t < T; t += NTHR) msum += (lmask[b * T + t] != 0);
    redA[tid] = msum;
    redB[tid] = (tid < LMAX) ? (labels[b * LMAX + tid] != 0) : 0;
    __syncthreads();

    if (tid == 0) {
        int tl = 0, ll = 0;
        for (int i = 0; i < NTHR; ++i) { tl += redA[i]; ll += redB[i]; }
        lens[0] = ll; lens[1] = tl;
    }
    if (tid < S_EXT) {
        int e2 = (tid >= 2) ? extLab[tid - 2] : 0;
        allowS[tid] = (es != 0) && (es != e2);
    }
    if (tid < 2) { alphaBuf[0][tid] = NEGV; alphaBuf[1][tid] = NEGV; }
    __syncthreads();
    const int labels_len = lens[0];
    const int logits_len = lens[1];

    // ---- async tile loader: 16 rows x 64 f32 = 256 x b128 chunks ----
    auto issueTile = [&](int d) {
        const float* src = logits + ((size_t)b * T + (size_t)d * TILE_T) * VDIM;
        float* dstBase = &ring[d & 1][0];
        for (int c = tid; c < (TILE_T * VDIM) / 4; c += NTHR) {
            int rr = c >> 4, cc = c & 15;
            unsigned voff = (unsigned)((rr * VDIM + cc * 4) * 4);
            unsigned ldsb = (unsigned)(uintptr_t)(dstBase + rr * ROWP + cc * 4);
            asm volatile("global_load_async_to_lds_b128 %0, %1, %2"
                         :: "v"(ldsb), "v"(voff), "s"(src) : "memory");
        }
    };

    issueTile(0);

    float* aP = alphaBuf[0];
    float* aN = alphaBuf[1];
    const int nTiles = T / TILE_T;

    // WMMA one-hot rows for this wave's two s-tiles.
    // 10 tiles over 5 waves; tile 9 is dead: extLab[144..159] == -1 -> A == 0,
    // so the chain is defined (D accumulates zeros) and every wave executes
    // both WMMA chains with EXEC all-ones (ISA requirement). Only the stores
    // are guarded.
    const int ts0 = wave, ts1 = wave + 5;
    const int es0 = extLab[ts0 * 16 + (lane & 15)];
    const int es1 = extLab[ts1 * 16 + (lane & 15)];
    const int kb  = (lane >= 16) ? 2 : 0;    // A/B 16x4 f32 layout: upper half-wave holds K=2,3
    const int nn  = lane & 15;               // N (time-within-tile) column

    for (int d = 0; d < nTiles; ++d) {
        asm volatile("s_wait_asynccnt 0x0" ::: "memory");  // tile d resident (this wave's loads)
        __syncthreads();                                   // all waves' data visible; prev DP done
        if (d + 1 < nTiles) issueTile(d + 1);              // prefetch next tile into other buffer

        const float* tb = &ring[d & 1][0];

        // per-row log-sum-exp (16 rows on lanes 0..15 of wave 0; off critical path)
        if (tid < TILE_T) {
            float m = NEGV;
            for (int j = 0; j < VDIM; ++j) m = fmaxf(m, tb[tid * ROWP + j]);
            float ss = 0.f;
            for (int j = 0; j < VDIM; ++j) ss += __expf(tb[tid * ROWP + j] - m);
            lseRow[tid] = m + __logf(ss);
        }

        // em tile = onehot(ext) x logits_tile  (exact fp32 gather on matrix pipe)
        v8f c0 = {}; v8f c1 = {};
        for (int k = 0; k < 16; ++k) {
            v2f bb; bb.x = tb[nn * ROWP + 4 * k + kb];
                    bb.y = tb[nn * ROWP + 4 * k + kb + 1];
            v2f a0; a0.x = (es0 == 4 * k + kb)     ? 1.f : 0.f;
                    a0.y = (es0 == 4 * k + kb + 1) ? 1.f : 0.f;
            v2f a1; a1.x = (es1 == 4 * k + kb)     ? 1.f : 0.f;
                    a1.y = (es1 == 4 * k + kb + 1) ? 1.f : 0.f;
            c0 = __builtin_amdgcn_wmma_f32_16x16x4_f32(false, a0, false, bb,
                                                       (short)0, c0, false, false);
            c1 = __builtin_amdgcn_wmma_f32_16x16x4_f32(false, a1, false, bb,
                                                       (short)0, c1, false, false);
        }
        __syncthreads();   // lseRow ready before writeback

        {   // D 16x16 layout: VGPR r -> M=r (lanes 0..15) / M=r+8 (lanes 16..31)
            const float ls   = lseRow[nn];
            const int   mOff = (lane >= 16) ? 8 : 0;
            #pragma unroll
            for (int r = 0; r < 8; ++r)
                em[nn * EMP + ts0 * 16 + mOff + r] = c0[r] - ls;
            if (ts1 < 9) {                                 // stores only; EXEC-masking OK
                #pragma unroll
                for (int r = 0; r < 8; ++r)
                    em[nn * EMP + ts1 * 16 + mOff + r] = c1[r] - ls;
            }
        }
        __syncthreads();

        // ---- 16 sequential DP steps (ping-pong alpha, 1 barrier/step) ----
        for (int tt = 0; tt < TILE_T; ++tt) {
            const int t = d * TILE_T + tt;
            if (tid < S_EXT) {
                float v;
                if (t == 0) {
                    v = (tid <= 1) ? em[tid] : NEGV;         // alpha0
                } else if (t < logits_len) {
                    float x0 = aP[tid + 2];
                    float x1 = aP[tid + 1];
                    float x2 = allowS[tid] ? aP[tid] : NEGV;
                    float m  = fmaxf(fmaxf(x0, x1), x2);
                    v = em[tt * EMP + tid] + m +
                        __logf(__expf(x0 - m) + __expf(x1 - m) + __expf(x2 - m));
                } else {
                    v = aP[tid + 2];                          // frozen past logits_len
                }
                aN[tid + 2] = v;
            }
            __syncthreads();
            float* tmp = aP; aP = aN; aN = tmp;
        }
    }

    if (tid == 0) {
        int   e   = 2 * labels_len;
        float ae  = aP[e + 2];
        float ae1 = aP[(e > 0 ? e - 1 : 0) + 2];
        float m   = fmaxf(ae, ae1);
        float loss = -(m + __logf(__expf(ae - m) + __expf(ae1 - m)));
        loss_out[b] = (labels_len <= logits_len) ? loss : 0.f;
    }
}

__global__ __launch_bounds__(256)
void ctc_reduce_mean(const float* __restrict__ lossb, float* __restrict__ out, int B)
{
    __shared__ float sm[256];
    float a = 0.f;
    for (int i = threadIdx.x; i < B; i += 256) a += lossb[i];
    sm[threadIdx.x] = a;
    __syncthreads();
    for (int st = 128; st > 0; st >>= 1) {
        if (threadIdx.x < st) sm[threadIdx.x] += sm[threadIdx.x + st];
        __syncthreads();
    }
    if (threadIdx.x == 0) out[0] = sm[0] / (float)B;
}

extern "C" void kernel_launch(void* const* d_in, const int* in_sizes, int n_in,
                              void* d_out, int out_size, void* d_ws, size_t ws_size,
                              hipStream_t stream)
{
    const int*   labels = (const int*)d_in[0];    // [B,64]
    const float* logits = (const float*)d_in[1];  // [B,T,64] fp32
    const int*   lmask  = (const int*)d_in[2];    // [B,T] 0/1
    const int B = in_sizes[0] / LMAX;             // 256
    const int T = in_sizes[2] / B;                // 1024

    float* lossb = (float*)d_ws;                  // B floats of scratch

    hipLaunchKernelGGL(ctc_dp_kernel, dim3(B), dim3(NTHR), 0, stream,
                       labels, logits, lmask, lossb, T);
    hipLaunchKernelGGL(ctc_reduce_mean, dim3(1), dim3(256), 0, stream,
                       lossb, (float*)d_out, B);
}